// MultiBoxLoss_326417515356
// MI455X (gfx1250) — compile-verified
//
#include <hip/hip_runtime.h>
#include <hip/hip_bf16.h>
#include <stdint.h>

#define BB 128
#define PP 8732
#define CC 21
#define OO 16
#define SLICES 8
#define WAVES 8
#define NCHUNK ((PP + 15) / 16)   // 546

typedef float v2f __attribute__((ext_vector_type(2)));
typedef float v8f __attribute__((ext_vector_type(8)));

// ---------------------------------------------------------------------------
// Kernel 1: SSD matching per image. One block per image.
// Produces conf_t[B,P], per-image smooth-L1 sum, per-image positive count.
// ---------------------------------------------------------------------------
__global__ __launch_bounds__(256) void k_match(
    const float* __restrict__ loc_data,   // [B,P,4]
    const float* __restrict__ dbox,       // [P,4] center form
    const float* __restrict__ targets,    // [B,O,5]
    int*   __restrict__ conf_t,           // [B,P]
    float* __restrict__ lossl_part,       // [B]
    int*   __restrict__ numpos)           // [B]
{
    __shared__ float s_truth[OO][5];
    __shared__ float s_area[OO];
    __shared__ unsigned long long s_best[OO];
    __shared__ float s_bto[PP];
    __shared__ unsigned char s_bti[PP];
    __shared__ float s_redf[256];
    __shared__ int   s_redi[256];

    const int b = blockIdx.x;
    const int tid = threadIdx.x;

    if (tid < OO * 5) ((float*)s_truth)[tid] = targets[(size_t)b * OO * 5 + tid];
    if (tid < OO) s_best[tid] = 0ull;
    __syncthreads();
    if (tid < OO)
        s_area[tid] = (s_truth[tid][2] - s_truth[tid][0]) *
                      (s_truth[tid][3] - s_truth[tid][1]);
    __syncthreads();

    unsigned long long best[OO];
#pragma unroll
    for (int o = 0; o < OO; ++o) best[o] = 0ull;

    for (int p = tid; p < PP; p += 256) {
        float4 d = ((const float4*)dbox)[p];
        float px0 = d.x - d.z * 0.5f, py0 = d.y - d.w * 0.5f;
        float px1 = d.x + d.z * 0.5f, py1 = d.y + d.w * 0.5f;
        float ap = d.z * d.w;
        float bmax = -1.0f; int bidx = 0;
#pragma unroll
        for (int o = 0; o < OO; ++o) {
            float ix = fminf(s_truth[o][2], px1) - fmaxf(s_truth[o][0], px0);
            float iy = fminf(s_truth[o][3], py1) - fmaxf(s_truth[o][1], py0);
            ix = fmaxf(ix, 0.0f); iy = fmaxf(iy, 0.0f);
            float inter = ix * iy;
            float ov = inter / (s_area[o] + ap - inter);
            if (ov > bmax) { bmax = ov; bidx = o; }   // first-max (smallest o)
            // pack (ov_bits, ~p): max() -> argmax with smallest-p tie-break
            unsigned long long pk =
                ((unsigned long long)__float_as_uint(ov) << 32) |
                (unsigned)(~(unsigned)p);
            best[o] = pk > best[o] ? pk : best[o];
        }
        s_bto[p] = bmax;
        s_bti[p] = (unsigned char)bidx;
    }
#pragma unroll
    for (int o = 0; o < OO; ++o)
        atomicMax(&s_best[o], best[o]);
    __syncthreads();

    if (tid == 0) {
#pragma unroll
        for (int o = 0; o < OO; ++o) {
            unsigned bp = ~(unsigned)(s_best[o] & 0xFFFFFFFFull);
            s_bto[bp] = 2.0f;
        }
        for (int o = 0; o < OO; ++o) {            // sequential: later o wins
            unsigned bp = ~(unsigned)(s_best[o] & 0xFFFFFFFFull);
            s_bti[bp] = (unsigned char)o;
        }
    }
    __syncthreads();

    float lsum = 0.0f; int np = 0;
    for (int p = tid; p < PP; p += 256) {
        int ti = s_bti[p];
        int lab = (int)s_truth[ti][4];
        int cf = (s_bto[p] < 0.5f) ? 0 : (lab + 1);
        conf_t[(size_t)b * PP + p] = cf;
        if (cf > 0) {
            ++np;
            float4 d = ((const float4*)dbox)[p];
            float mx0 = s_truth[ti][0], my0 = s_truth[ti][1];
            float mx1 = s_truth[ti][2], my1 = s_truth[ti][3];
            float gcx = ((mx0 + mx1) * 0.5f - d.x) / (0.1f * d.z);
            float gcy = ((my0 + my1) * 0.5f - d.y) / (0.1f * d.w);
            float gw  = __logf((mx1 - mx0) / d.z) * (1.0f / 0.2f);
            float gh  = __logf((my1 - my0) / d.w) * (1.0f / 0.2f);
            float4 l = ((const float4*)loc_data)[(size_t)b * PP + p];
            float ad, t;
            ad = fabsf(l.x - gcx); t = (ad < 1.0f) ? 0.5f * ad * ad : ad - 0.5f; lsum += t;
            ad = fabsf(l.y - gcy); t = (ad < 1.0f) ? 0.5f * ad * ad : ad - 0.5f; lsum += t;
            ad = fabsf(l.z - gw ); t = (ad < 1.0f) ? 0.5f * ad * ad : ad - 0.5f; lsum += t;
            ad = fabsf(l.w - gh ); t = (ad < 1.0f) ? 0.5f * ad * ad : ad - 0.5f; lsum += t;
        }
    }
    s_redf[tid] = lsum; s_redi[tid] = np;
    __syncthreads();
    for (int s = 128; s > 0; s >>= 1) {
        if (tid < s) { s_redf[tid] += s_redf[tid + s]; s_redi[tid] += s_redi[tid + s]; }
        __syncthreads();
    }
    if (tid == 0) { lossl_part[b] = s_redf[0]; numpos[b] = s_redi[0]; }
}

// ---------------------------------------------------------------------------
// Kernel 2: per-prior cross entropy via WMMA logsumexp row-sums.
// Grid (B, SLICES). Each wave handles 16 priors per chunk; conf tiles are
// staged global->LDS with async-to-LDS loads (double buffered). The per-lane
// element path is fully branchless: batched ds loads -> one wait, masked with
// cndmask, row max completed by a single xor-16 shuffle across the lane pair.
// ---------------------------------------------------------------------------
__global__ __launch_bounds__(256) void k_ce(
    const float* __restrict__ conf_data,  // [B,P,C]
    const int*   __restrict__ conf_t,     // [B,P]
    float* __restrict__ ce_neg,           // [B,P]
    float* __restrict__ cepos_part)       // [B*SLICES]
{
    __shared__ float s_tile[WAVES][2][16][24];  // row padded to 24 (cols 21..23 junk)
    __shared__ float s_sum[WAVES][16];
    __shared__ float s_redf[256];

    const int b    = blockIdx.x;
    const int tid  = threadIdx.x;
    const int w    = tid >> 5;
    const int lane = tid & 31;
    const int r    = lane & 15;   // matrix row this lane owns
    const int hi   = lane >> 4;   // K-half selector (A-matrix 16x4 f32 layout)

    const float* cbase = conf_data + (size_t)b * PP * CC;
    float acc_pos = 0.0f;
    int buf = 0;
    const float NEG = -1.0e30f;

    for (int c = blockIdx.y * WAVES + w; c < NCHUNK; c += SLICES * WAVES) {
        const int p0 = c * 16;
        const int rv = min(16, PP - p0);
        const int nelem = rv * CC;
        const unsigned lds_base = (unsigned)(uintptr_t)&s_tile[w][buf][0][0];

        // --- async stage of the 16x21 f32 tile into LDS ---
        for (int e = lane; e < nelem; e += 32) {
            int row = e / CC;
            int col = e - row * CC;
            unsigned lds = lds_base + (unsigned)((row * 24 + col) * 4);
            unsigned long long g =
                (unsigned long long)(uintptr_t)(cbase + (size_t)(p0 + row) * CC + col);
            asm volatile("global_load_async_to_lds_b32 %0, %1, off"
                         :: "v"(lds), "v"(g) : "memory");
        }
        // prefetch the next chunk this wave will touch
        {
            int pn = p0 + SLICES * WAVES * 16;
            if (pn < PP) __builtin_prefetch(cbase + (size_t)min(pn + r, PP - 1) * CC, 0, 1);
        }
        asm volatile("s_wait_asynccnt 0" ::: "memory");
        __builtin_amdgcn_wave_barrier();

        const bool rowok = r < rv;

        // batched unconditional LDS reads (pairs are 8B aligned -> ds_load_b64)
        float x[12];
#pragma unroll
        for (int kk = 0; kk < 6; ++kk) {
            int c0 = kk * 4 + hi * 2;
            x[2 * kk]     = s_tile[w][buf][r][c0];
            x[2 * kk + 1] = s_tile[w][buf][r][c0 + 1];
        }
        // mask invalid elements (cols >= 21, or dead tail rows) via cndmask
#pragma unroll
        for (int kk = 0; kk < 6; ++kk) {
            int c0 = kk * 4 + hi * 2;
            if (!(rowok && (c0     < CC))) x[2 * kk]     = NEG;
            if (!(rowok && (c0 + 1 < CC))) x[2 * kk + 1] = NEG;
        }
        // row max: reduce own 12 then complete across the lane pair
        float mv = NEG;
#pragma unroll
        for (int i = 0; i < 12; ++i) mv = fmaxf(mv, x[i]);
        mv = fmaxf(mv, __shfl_xor(mv, 16));

        // --- row-sum of exp(x - max) over 21 classes via 6 chained WMMAs ---
        v8f acc = {};
        v2f ones; ones.x = 1.0f; ones.y = 1.0f;
#pragma unroll
        for (int kk = 0; kk < 6; ++kk) {
            v2f a;
            a.x = __expf(x[2 * kk]     - mv);   // masked -> exp(-huge) == 0
            a.y = __expf(x[2 * kk + 1] - mv);
            acc = __builtin_amdgcn_wmma_f32_16x16x4_f32(
                false, a, false, ones, (short)0, acc, false, false);
        }
        // D layout: VGPR j, lanes 0-15 -> M=j ; lanes 16-31 -> M=j+8
        if (lane == 0) {
#pragma unroll
            for (int i = 0; i < 8; ++i) s_sum[w][i] = acc[i];
        }
        if (lane == 16) {
#pragma unroll
            for (int i = 0; i < 8; ++i) s_sum[w][8 + i] = acc[i];
        }
        __builtin_amdgcn_wave_barrier();

        if (lane < rv) {
            int p = p0 + lane;
            int t = conf_t[(size_t)b * PP + p];
            float xt = s_tile[w][buf][lane][t];
            float ce = mv + __logf(s_sum[w][lane]) - xt;   // logsumexp - x_t
            bool pos = t > 0;
            ce_neg[(size_t)b * PP + p] = pos ? 0.0f : ce;
            if (pos) acc_pos += ce;
        }
        __builtin_amdgcn_wave_barrier();
        buf ^= 1;
    }

    s_redf[tid] = acc_pos;
    __syncthreads();
    for (int s = 128; s > 0; s >>= 1) {
        if (tid < s) s_redf[tid] += s_redf[tid + s];
        __syncthreads();
    }
    if (tid == 0) cepos_part[b * SLICES + blockIdx.y] = s_redf[0];
}

// ---------------------------------------------------------------------------
// Kernel 3: top-K selection per image via binary search on float bit space.
// Sum over top-K of ce_neg is tie-invariant => equals reference rank test.
// ---------------------------------------------------------------------------
__global__ __launch_bounds__(256) void k_topk(
    const float* __restrict__ ce_neg,  // [B,P]
    const int*   __restrict__ numpos,  // [B]
    float* __restrict__ topk)          // [B]
{
    __shared__ float s_ce[PP];
    __shared__ unsigned s_cnt[256];
    __shared__ float s_redf[256];

    const int b = blockIdx.x;
    const int tid = threadIdx.x;

    for (int i = tid; i < PP; i += 256) s_ce[i] = ce_neg[(size_t)b * PP + i];
    __syncthreads();

    const int np = numpos[b];
    const unsigned K = (unsigned)min(np * 3, PP);
    if (K == 0) { if (tid == 0) topk[b] = 0.0f; return; }

    // smallest x with count(bits > x) < K  ==  bit pattern of K-th largest
    unsigned lo = 0u, hi = 0x7F800000u;
    while (lo < hi) {
        unsigned mid = (lo + hi) >> 1;
        unsigned cnt = 0;
        for (int i = tid; i < PP; i += 256)
            cnt += (__float_as_uint(s_ce[i]) > mid) ? 1u : 0u;
        s_cnt[tid] = cnt;
        __syncthreads();
        for (int s = 128; s > 0; s >>= 1) {
            if (tid < s) s_cnt[tid] += s_cnt[tid + s];
            __syncthreads();
        }
        unsigned tot = s_cnt[0];
        __syncthreads();
        if (tot >= K) lo = mid + 1; else hi = mid;
    }
    const unsigned T = lo;
    const float Tv = __uint_as_float(T);

    float ssum = 0.0f; unsigned cgt = 0;
    for (int i = tid; i < PP; i += 256) {
        float v = s_ce[i];
        if (__float_as_uint(v) > T) { ssum += v; ++cgt; }
    }
    s_redf[tid] = ssum; s_cnt[tid] = cgt;
    __syncthreads();
    for (int s = 128; s > 0; s >>= 1) {
        if (tid < s) { s_redf[tid] += s_redf[tid + s]; s_cnt[tid] += s_cnt[tid + s]; }
        __syncthreads();
    }
    if (tid == 0)
        topk[b] = s_redf[0] + (float)(K - s_cnt[0]) * Tv;
}

// ---------------------------------------------------------------------------
// Kernel 4: fixed-order final combine (deterministic).
// ---------------------------------------------------------------------------
__global__ void k_final(const float* __restrict__ lossl_part,
                        const int*   __restrict__ numpos,
                        const float* __restrict__ cepos_part,
                        const float* __restrict__ topk,
                        float* __restrict__ out)
{
    if (threadIdx.x == 0 && blockIdx.x == 0) {
        float ll = 0.0f, lc = 0.0f; int n = 0;
        for (int b = 0; b < BB; ++b) { ll += lossl_part[b]; n += numpos[b]; lc += topk[b]; }
        for (int i = 0; i < BB * SLICES; ++i) lc += cepos_part[i];
        float N = (float)n;
        out[0] = ll / N;
        out[1] = lc / N;
    }
}

// ---------------------------------------------------------------------------
extern "C" void kernel_launch(void* const* d_in, const int* in_sizes, int n_in,
                              void* d_out, int out_size, void* d_ws, size_t ws_size,
                              hipStream_t stream) {
    (void)in_sizes; (void)n_in; (void)out_size; (void)ws_size;
    const float* loc_data  = (const float*)d_in[0];
    const float* conf_data = (const float*)d_in[1];
    const float* dbox_list = (const float*)d_in[2];
    const float* targets   = (const float*)d_in[3];
    float* out = (float*)d_out;

    char* ws = (char*)d_ws;
    const size_t BP4 = (size_t)BB * PP * 4;                 // 4,470,784 (16B aligned)
    int*   conf_t     = (int*)  (ws);
    float* ce_neg     = (float*)(ws + BP4);
    float* lossl_part = (float*)(ws + 2 * BP4);
    int*   numpos     = (int*)  (ws + 2 * BP4 + 512);
    float* cepos_part = (float*)(ws + 2 * BP4 + 1024);
    float* topk       = (float*)(ws + 2 * BP4 + 1024 + (size_t)BB * SLICES * 4);

    k_match<<<BB, 256, 0, stream>>>(loc_data, dbox_list, targets,
                                    conf_t, lossl_part, numpos);
    k_ce<<<dim3(BB, SLICES), 256, 0, stream>>>(conf_data, conf_t, ce_neg, cepos_part);
    k_topk<<<BB, 256, 0, stream>>>(ce_neg, numpos, topk);
    k_final<<<1, 32, 0, stream>>>(lossl_part, numpos, cepos_part, topk, out);
}